// NeighborhoodAttentionAggregator_6923487282080
// MI455X (gfx1250) — compile-verified
//
#include <hip/hip_runtime.h>
#include <hip/hip_bf16.h>
#include <math.h>

typedef __attribute__((ext_vector_type(2))) float v2f;
typedef __attribute__((ext_vector_type(4))) float v4f;
typedef __attribute__((ext_vector_type(8))) float v8f;

#define DIM_C   256
#define ADIM_C  64
#define KN_C    16
#define SCALE_C 0.125f   // 64^-0.5
#define LNEPS_C 1e-5f

#define BM   128   // block M tile (8 waves x 16)
#define BN   64    // block N tile
#define BK   16    // K chunk (4 WMMA k-steps)
#define LDST 20    // padded LDS row stride (floats): 16B-aligned, bank-conflict-free

// ---------------------------------------------------------------------------
// C[M, Nc] = A[M, Kd] @ W[Nc, Kd]^T  (fp32 exact, WMMA f32 16x16x4)
// 256-thread block -> 128x64 C tile, LDS double-buffered over K chunks of 16.
// Global loads for chunk c+1 are issued before the LDS compute of chunk c,
// so WMMAs wait only on short-latency DScnt, not on L2 loads.
// Fragment layout (ISA 7.12.2): lane (m=lane&15, h=lane>>4) holds
//   A[m][k0+2h], A[m][k0+2h+1] (one b64); B symmetric with N; C/D v8f.
// Requires: M % 16 == 0, Kd % 16 == 0, Nc == gridDim.y * 64.
// ---------------------------------------------------------------------------
__global__ __launch_bounds__(256)
void wmma_gemm_nt_f32(const float* __restrict__ A,
                      const float* __restrict__ W,
                      float* __restrict__ C,
                      int M, int Nc, int Kd)
{
    __shared__ __align__(16) float sA[2][BM * LDST];
    __shared__ __align__(16) float sW[2][BN * LDST];

    const int tid  = threadIdx.x;
    const int lane = tid & 31;
    const int wv   = tid >> 5;           // wave 0..7
    const int m    = lane & 15;
    const int h    = lane >> 4;          // selects K pair {0,1} vs {2,3}

    const int row0B = blockIdx.x * BM;
    const int n0    = blockIdx.y * BN;

    // cooperative global->LDS load mapping
    const int ar  = tid >> 1;            // 0..127 : A row within tile
    const int ac0 = (tid & 1) * 8;       // 0 or 8 : col offset within chunk
    const int arow_g = min(row0B + ar, M - 1);            // clamp (branch-free)
    const float* aptr = A + (size_t)arow_g * Kd + ac0;

    const int wr  = tid >> 2;            // 0..63 : W row within tile
    const int wc0 = (tid & 3) * 4;       // 0,4,8,12
    const float* wptr = W + (size_t)(n0 + wr) * Kd + wc0;

    const int NCHUNK = Kd / BK;

    v8f acc0 = {}, acc1 = {}, acc2 = {}, acc3 = {};

    // ---- prologue: chunk 0 -> buffer 0
    {
        v4f a0 = *(const v4f*)(aptr + 0);
        v4f a1 = *(const v4f*)(aptr + 4);
        v4f w0 = *(const v4f*)(wptr + 0);
        *(v4f*)(&sA[0][ar * LDST + ac0 + 0]) = a0;
        *(v4f*)(&sA[0][ar * LDST + ac0 + 4]) = a1;
        *(v4f*)(&sW[0][wr * LDST + wc0])     = w0;
    }
    __syncthreads();

    for (int c = 0; c < NCHUNK; ++c) {
        const int cur = c & 1;
        const int nxt = cur ^ 1;
        const bool have_next = (c + 1 < NCHUNK);

        // issue next chunk's global loads before computing current chunk
        v4f a0, a1, w0;
        if (have_next) {
            const int kc = (c + 1) * BK;
            a0 = *(const v4f*)(aptr + kc + 0);
            a1 = *(const v4f*)(aptr + kc + 4);
            w0 = *(const v4f*)(wptr + kc);
        }

        // compute current chunk from LDS (4 k-steps x 4 N-tiles)
        const float* As = &sA[cur][(wv * 16 + m) * LDST + 2 * h];
        const float* W0 = &sW[cur][( 0 + m) * LDST + 2 * h];
        const float* W1 = &sW[cur][(16 + m) * LDST + 2 * h];
        const float* W2 = &sW[cur][(32 + m) * LDST + 2 * h];
        const float* W3 = &sW[cur][(48 + m) * LDST + 2 * h];

        #pragma unroll
        for (int kk = 0; kk < 4; ++kk) {
            v2f a  = *(const v2f*)(As + kk * 4);
            v2f b0 = *(const v2f*)(W0 + kk * 4);
            v2f b1 = *(const v2f*)(W1 + kk * 4);
            v2f b2 = *(const v2f*)(W2 + kk * 4);
            v2f b3 = *(const v2f*)(W3 + kk * 4);
            acc0 = __builtin_amdgcn_wmma_f32_16x16x4_f32(false, a, false, b0, (short)0, acc0, false, false);
            acc1 = __builtin_amdgcn_wmma_f32_16x16x4_f32(false, a, false, b1, (short)0, acc1, false, false);
            acc2 = __builtin_amdgcn_wmma_f32_16x16x4_f32(false, a, false, b2, (short)0, acc2, false, false);
            acc3 = __builtin_amdgcn_wmma_f32_16x16x4_f32(false, a, false, b3, (short)0, acc3, false, false);
        }

        // park next chunk into the other LDS buffer
        if (have_next) {
            *(v4f*)(&sA[nxt][ar * LDST + ac0 + 0]) = a0;
            *(v4f*)(&sA[nxt][ar * LDST + ac0 + 4]) = a1;
            *(v4f*)(&sW[nxt][wr * LDST + wc0])     = w0;
        }
        __syncthreads();
    }

    // ---- store C (wave-uniform guard keeps EXEC behavior trivial; WMMAs done)
    const int row_w0 = row0B + wv * 16;
    if (row_w0 < M) {
        const int n  = lane & 15;
        const int mh = lane >> 4;
        #pragma unroll
        for (int i = 0; i < 8; ++i) {
            const size_t r = (size_t)(row_w0 + i + 8 * mh) * Nc + n0 + n;
            C[r +  0] = acc0[i];
            C[r + 16] = acc1[i];
            C[r + 32] = acc2[i];
            C[r + 48] = acc3[i];
        }
    }
}

// ---------------------------------------------------------------------------
// Per-query attention + gate + residual + LayerNorm.
// One 256-thread block (8 wave32) per query. K_all/V_all gathers hit L2
// (V_all = 20 MB, K_all = 5 MB, both resident in the 192 MB L2).
// ---------------------------------------------------------------------------
__global__ __launch_bounds__(256)
void neigh_attn_kernel(const float* __restrict__ query,   // [B,256]
                       const float* __restrict__ qproj,   // [B,64]
                       const float* __restrict__ Kall,    // [N,64]
                       const float* __restrict__ Vall,    // [N,256]
                       const int*   __restrict__ nidx,    // [B,16]
                       const float* __restrict__ nw,      // [B,16]
                       const float* __restrict__ Wg,      // [256]
                       const float* __restrict__ bg,      // [1]
                       const float* __restrict__ gamma,   // [256]
                       const float* __restrict__ beta,    // [256]
                       float* __restrict__ out)           // [B,256]
{
    __shared__ int   s_idx[KN_C];
    __shared__ float s_w[KN_C];
    __shared__ float s_sc[KN_C];
    __shared__ float s_e[KN_C];
    __shared__ float s_red[8];
    __shared__ float s_red2[8];

    const int b   = blockIdx.x;
    const int tid = threadIdx.x;

    if (tid < KN_C) {
        s_idx[tid] = nidx[(size_t)b * KN_C + tid];
        s_w[tid]   = nw[(size_t)b * KN_C + tid];
    }
    __syncthreads();

    // ---- scores: 16 groups of 16 lanes, each group dots q (64) with K_all row
    {
        const int j = tid >> 4;          // neighbor 0..15
        const int l = tid & 15;          // lane within group
        const float* qv = qproj + (size_t)b * ADIM_C;
        const float* kv = Kall + (size_t)s_idx[j] * ADIM_C;
        float p = 0.f;
        #pragma unroll
        for (int t = 0; t < 4; ++t) p += qv[l * 4 + t] * kv[l * 4 + t];
        p += __shfl_xor(p, 8, 16);
        p += __shfl_xor(p, 4, 16);
        p += __shfl_xor(p, 2, 16);
        p += __shfl_xor(p, 1, 16);
        if (l == 0) s_sc[j] = p * SCALE_C;
    }
    __syncthreads();

    // ---- confidence bias (lanes 0..15)
    if (tid < KN_C) {
        float wsum = 0.f;
        #pragma unroll
        for (int j = 0; j < KN_C; ++j) wsum += s_w[j];
        s_sc[tid] = s_sc[tid] + logf(s_w[tid] / (wsum + 1e-6f) + 1e-6f);
    }
    __syncthreads();

    // ---- softmax numerator (lanes 0..15)
    if (tid < KN_C) {
        float mx = -3.402823e38f;
        #pragma unroll
        for (int j = 0; j < KN_C; ++j) mx = fmaxf(mx, s_sc[j]);
        s_e[tid] = expf(s_sc[tid] - mx);
    }
    __syncthreads();

    float esum = 0.f;
    #pragma unroll
    for (int j = 0; j < KN_C; ++j) esum += s_e[j];
    const float inv_esum = 1.f / esum;

    // ---- context: each thread owns one of 256 dims; 16 coalesced gathered rows
    const int d = tid;
    float ctx = 0.f;
    #pragma unroll
    for (int j = 0; j < KN_C; ++j)
        ctx += s_e[j] * Vall[(size_t)s_idx[j] * DIM_C + d];
    ctx *= inv_esum;

    const float qd = query[(size_t)b * DIM_C + d];

    // ---- gate = sigmoid(query . Wg + bg): block reduction
    float g = qd * Wg[d];
    g += __shfl_xor(g, 16, 32);
    g += __shfl_xor(g,  8, 32);
    g += __shfl_xor(g,  4, 32);
    g += __shfl_xor(g,  2, 32);
    g += __shfl_xor(g,  1, 32);
    const int wid = tid >> 5;
    if ((tid & 31) == 0) s_red[wid] = g;
    __syncthreads();
    float gdot = 0.f;
    #pragma unroll
    for (int w = 0; w < 8; ++w) gdot += s_red[w];
    const float gate = 1.f / (1.f + expf(-(gdot + bg[0])));

    const float x = qd + gate * ctx;

    // ---- LayerNorm: block reductions of sum and sumsq
    __syncthreads();                      // s_red reuse safety
    float s1 = x, s2 = x * x;
    s1 += __shfl_xor(s1, 16, 32);  s2 += __shfl_xor(s2, 16, 32);
    s1 += __shfl_xor(s1,  8, 32);  s2 += __shfl_xor(s2,  8, 32);
    s1 += __shfl_xor(s1,  4, 32);  s2 += __shfl_xor(s2,  4, 32);
    s1 += __shfl_xor(s1,  2, 32);  s2 += __shfl_xor(s2,  2, 32);
    s1 += __shfl_xor(s1,  1, 32);  s2 += __shfl_xor(s2,  1, 32);
    if ((tid & 31) == 0) { s_red[wid] = s1; s_red2[wid] = s2; }
    __syncthreads();
    float t1 = 0.f, t2 = 0.f;
    #pragma unroll
    for (int w = 0; w < 8; ++w) { t1 += s_red[w]; t2 += s_red2[w]; }
    const float mean = t1 * (1.f / DIM_C);
    const float var  = t2 * (1.f / DIM_C) - mean * mean;
    const float rstd = rsqrtf(var + LNEPS_C);

    out[(size_t)b * DIM_C + d] = (x - mean) * rstd * gamma[d] + beta[d];
}

// ---------------------------------------------------------------------------
extern "C" void kernel_launch(void* const* d_in, const int* in_sizes, int n_in,
                              void* d_out, int out_size, void* d_ws, size_t ws_size,
                              hipStream_t stream)
{
    const float* query   = (const float*)d_in[0];   // [B,256]
    const float* all_emb = (const float*)d_in[1];   // [N,256]
    const int*   nidx    = (const int*)  d_in[2];   // [B,16]
    const float* nw      = (const float*)d_in[3];   // [B,16]
    const float* Wq      = (const float*)d_in[4];   // [64,256]
    const float* Wk      = (const float*)d_in[5];   // [64,256]
    const float* Wv      = (const float*)d_in[6];   // [256,256]
    const float* Wg      = (const float*)d_in[7];   // [1,256]
    const float* bg      = (const float*)d_in[8];   // [1]
    const float* gamma   = (const float*)d_in[9];   // [256]
    const float* beta    = (const float*)d_in[10];  // [256]
    float* out = (float*)d_out;

    const int B = in_sizes[0] / DIM_C;   // 65536
    const int N = in_sizes[1] / DIM_C;   // 20000

    // Workspace: qproj [B,64] | Kall [N,64] | Vall [N,256]  (~42 MB)
    char* ws = (char*)d_ws;
    float* qproj = (float*)ws;
    float* Kall  = (float*)(ws + (size_t)B * ADIM_C * sizeof(float));
    float* Vall  = (float*)(ws + (size_t)B * ADIM_C * sizeof(float)
                               + (size_t)N * ADIM_C * sizeof(float));

    const dim3 blk(256);
    const int gB = (B + BM - 1) / BM;    // 512
    const int gN = (N + BM - 1) / BM;    // 157

    // q = query @ Wq^T        : [B,64]
    wmma_gemm_nt_f32<<<dim3(gB, 1), blk, 0, stream>>>(query,   Wq, qproj, B, ADIM_C, DIM_C);
    // K_all = all_emb @ Wk^T  : [N,64]
    wmma_gemm_nt_f32<<<dim3(gN, 1), blk, 0, stream>>>(all_emb, Wk, Kall,  N, ADIM_C, DIM_C);
    // V_all = all_emb @ Wv^T  : [N,256]
    wmma_gemm_nt_f32<<<dim3(gN, 4), blk, 0, stream>>>(all_emb, Wv, Vall,  N, DIM_C,  DIM_C);

    // Fused attention + gate + residual + LayerNorm
    neigh_attn_kernel<<<dim3(B), dim3(256), 0, stream>>>(
        query, qproj, Kall, Vall, nidx, nw, Wg, bg, gamma, beta, out);
}